// AttentionalFlow_20083267076145
// MI455X (gfx1250) — compile-verified
//
#include <hip/hip_runtime.h>
#include <hip/hip_bf16.h>

typedef float v2f __attribute__((ext_vector_type(2)));
typedef float v8f __attribute__((ext_vector_type(8)));

#define B_  32
#define T_  512
#define J_  64
#define D_  512
#define TT  32          // T rows per workgroup

// ---------------------------------------------------------------------------
// Kernel 1: S = (ctx*wm) @ q^T + c_term[:,None] + q_term[None,:]
//           P = softmax_j(S)  (normalized, stored to ws)
//           M = max_j(S)      (stored to ws)
// grid: B * T/TT = 512 blocks, 256 threads (8 waves)
// ---------------------------------------------------------------------------
__global__ void __launch_bounds__(256)
k1_scores(const float* __restrict__ ctx, const float* __restrict__ qry,
          const float* __restrict__ W, float* __restrict__ P,
          float* __restrict__ M)
{
    __shared__ float s_wm[D_];
    __shared__ float s_ct[TT];
    __shared__ float s_qt[J_];
    __shared__ float s_S[TT * J_];

    const int tid = threadIdx.x;
    const int b   = blockIdx.x >> 4;          // 16 T-tiles per batch
    const int t0  = (blockIdx.x & 15) * TT;

    const float* wc = W;
    const float* wq = W + D_;
    const float* wm = W + 2 * D_;

    s_wm[tid]       = wm[tid];
    s_wm[tid + 256] = wm[tid + 256];
    if (tid < TT) s_ct[tid] = 0.f;
    if (tid < J_) s_qt[tid] = 0.f;
    __syncthreads();

    // c_term partials: 8 threads per row (32 rows), 64 elems each
    {
        const int row  = tid & 31;
        const int part = tid >> 5;
        const float* cr = ctx + ((size_t)(b * T_ + t0 + row)) * D_ + part * 64;
        const float* wr = wc + part * 64;
        float s = 0.f;
        #pragma unroll 8
        for (int i = 0; i < 64; ++i) s += cr[i] * wr[i];
        atomicAdd(&s_ct[row], s);
    }
    // q_term partials: 4 threads per row (64 rows), 128 elems each
    {
        const int row  = tid & 63;
        const int part = tid >> 6;
        const float* qr = qry + ((size_t)(b * J_ + row)) * D_ + part * 128;
        const float* wr = wq + part * 128;
        float s = 0.f;
        #pragma unroll 8
        for (int i = 0; i < 128; ++i) s += qr[i] * wr[i];
        atomicAdd(&s_qt[row], s);
    }
    __syncthreads();

    // WMMA: wave w computes the 16x16 tile at rows wt, cols wj
    const int w    = tid >> 5;
    const int lane = tid & 31;
    const int half = lane >> 4;
    const int lm   = lane & 15;
    const int wt   = (w >> 2) * 16;
    const int wj   = (w & 3) * 16;

    const float* arow = ctx + ((size_t)(b * T_ + t0 + wt + lm)) * D_;
    const float* brow = qry + ((size_t)(b * J_ + wj + lm)) * D_;

    v8f acc = {};
    #pragma unroll 4
    for (int k = 0; k < D_; k += 4) {
        const int kk = k + half * 2;
        v2f a, bq;
        a.x  = arow[kk]     * s_wm[kk];
        a.y  = arow[kk + 1] * s_wm[kk + 1];
        bq.x = brow[kk];
        bq.y = brow[kk + 1];
        acc = __builtin_amdgcn_wmma_f32_16x16x4_f32(false, a, false, bq,
                                                    (short)0, acc, false, false);
    }

    #pragma unroll
    for (int r = 0; r < 8; ++r) {
        const int rl = wt + half * 8 + r;   // local row (M)
        const int cl = wj + lm;             // local col (N)
        s_S[rl * J_ + cl] = acc[r] + s_ct[rl] + s_qt[cl];
    }
    __syncthreads();

    // softmax over j, one thread per row
    if (tid < TT) {
        float* srow = &s_S[tid * J_];
        float mx = srow[0];
        #pragma unroll
        for (int j = 1; j < J_; ++j) mx = fmaxf(mx, srow[j]);
        float sum = 0.f;
        #pragma unroll
        for (int j = 0; j < J_; ++j) { float e = __expf(srow[j] - mx); srow[j] = e; sum += e; }
        const float inv = 1.f / sum;
        float* prow = P + ((size_t)(b * T_ + t0 + tid)) * J_;
        #pragma unroll
        for (int j = 0; j < J_; ++j) prow[j] = srow[j] * inv;
        M[b * T_ + t0 + tid] = mx;
    }
}

// ---------------------------------------------------------------------------
// Kernel 2: bw = softmax_t(M[b,:]);  q2c[b,d] = sum_t bw[t]*ctx[b,t,d]
// grid: B blocks, 256 threads
// ---------------------------------------------------------------------------
__global__ void __launch_bounds__(256)
k2_q2c(const float* __restrict__ ctx, const float* __restrict__ M,
       float* __restrict__ q2c)
{
    __shared__ float s_w[T_];
    __shared__ float s_red[256];
    const int b   = blockIdx.x;
    const int tid = threadIdx.x;

    const float m0 = M[b * T_ + tid];
    const float m1 = M[b * T_ + tid + 256];
    s_red[tid] = fmaxf(m0, m1);
    __syncthreads();
    for (int s = 128; s > 0; s >>= 1) {
        if (tid < s) s_red[tid] = fmaxf(s_red[tid], s_red[tid + s]);
        __syncthreads();
    }
    const float mx = s_red[0];
    __syncthreads();

    const float e0 = __expf(m0 - mx);
    const float e1 = __expf(m1 - mx);
    s_w[tid]       = e0;
    s_w[tid + 256] = e1;
    s_red[tid]     = e0 + e1;
    __syncthreads();
    for (int s = 128; s > 0; s >>= 1) {
        if (tid < s) s_red[tid] += s_red[tid + s];
        __syncthreads();
    }
    const float inv = 1.f / s_red[0];

    float acc0 = 0.f, acc1 = 0.f;
    const float* cb = ctx + (size_t)b * T_ * D_;
    for (int t = 0; t < T_; ++t) {
        const float wt = s_w[t];
        acc0 += wt * cb[(size_t)t * D_ + tid];
        acc1 += wt * cb[(size_t)t * D_ + tid + 256];
    }
    q2c[b * D_ + tid]       = acc0 * inv;
    q2c[b * D_ + tid + 256] = acc1 * inv;
}

// ---------------------------------------------------------------------------
// Kernel 3: c2q = P @ q  (WMMA), then G = [ctx, c2q, ctx*c2q, ctx*q2c]
// grid: B * T/TT = 512 blocks, 256 threads (8 waves)
// ---------------------------------------------------------------------------
__global__ void __launch_bounds__(256)
k3_out(const float* __restrict__ ctx, const float* __restrict__ qry,
       const float* __restrict__ P, const float* __restrict__ q2c,
       float* __restrict__ out)
{
    __shared__ float s_c2q[TT * D_];   // 64 KB
    const int tid = threadIdx.x;
    const int b   = blockIdx.x >> 4;
    const int t0  = (blockIdx.x & 15) * TT;

    const int w    = tid >> 5;
    const int lane = tid & 31;
    const int half = lane >> 4;
    const int lm   = lane & 15;
    const int wt   = (w >> 2) * 16;      // row offset (2 row-tiles)
    const int n00  = (w & 3) * 128;      // 4 waves split the 512 cols

    const float* prow = P + ((size_t)(b * T_ + t0 + wt + lm)) * J_;
    const float* qb   = qry + (size_t)b * J_ * D_;

    for (int n0 = n00; n0 < n00 + 128; n0 += 16) {
        v8f acc = {};
        #pragma unroll
        for (int k = 0; k < J_; k += 4) {
            const int kb = k + half * 2;
            v2f a, bq;
            a.x  = prow[kb];
            a.y  = prow[kb + 1];
            bq.x = qb[(size_t)kb * D_ + n0 + lm];
            bq.y = qb[(size_t)(kb + 1) * D_ + n0 + lm];
            acc = __builtin_amdgcn_wmma_f32_16x16x4_f32(false, a, false, bq,
                                                        (short)0, acc, false, false);
        }
        #pragma unroll
        for (int r = 0; r < 8; ++r) {
            const int rl = wt + half * 8 + r;
            s_c2q[rl * D_ + n0 + lm] = acc[r];
        }
    }
    __syncthreads();

    const float qc0 = q2c[b * D_ + tid];
    const float qc1 = q2c[b * D_ + tid + 256];
    const float* cb = ctx + ((size_t)(b * T_ + t0)) * D_;
    float* ob = out + ((size_t)(b * T_ + t0)) * (4 * D_);
    for (int t = 0; t < TT; ++t) {
        const float cv0 = cb[(size_t)t * D_ + tid];
        const float cv1 = cb[(size_t)t * D_ + tid + 256];
        const float c20 = s_c2q[t * D_ + tid];
        const float c21 = s_c2q[t * D_ + tid + 256];
        float* orow = ob + (size_t)t * (4 * D_);
        orow[tid]              = cv0;
        orow[tid + 256]        = cv1;
        orow[D_ + tid]         = c20;
        orow[D_ + tid + 256]   = c21;
        orow[2*D_ + tid]       = cv0 * c20;
        orow[2*D_ + tid + 256] = cv1 * c21;
        orow[3*D_ + tid]       = cv0 * qc0;
        orow[3*D_ + tid + 256] = cv1 * qc1;
    }
}

// ---------------------------------------------------------------------------
extern "C" void kernel_launch(void* const* d_in, const int* in_sizes, int n_in,
                              void* d_out, int out_size, void* d_ws, size_t ws_size,
                              hipStream_t stream) {
    const float* ctx = (const float*)d_in[0];   // (32,512,512)
    const float* qry = (const float*)d_in[1];   // (32,64,512)
    const float* W   = (const float*)d_in[2];   // (1536,)
    float* out = (float*)d_out;                 // (32,512,2048)

    float* P   = (float*)d_ws;                          // B*T*J
    float* M   = P + (size_t)B_ * T_ * J_;              // B*T
    float* q2c = M + (size_t)B_ * T_;                   // B*D

    k1_scores<<<dim3(B_ * (T_ / TT)), dim3(256), 0, stream>>>(ctx, qry, W, P, M);
    k2_q2c  <<<dim3(B_),              dim3(256), 0, stream>>>(ctx, M, q2c);
    k3_out  <<<dim3(B_ * (T_ / TT)), dim3(256), 0, stream>>>(ctx, qry, P, q2c, out);
}